// Decoder_55310588838217
// MI455X (gfx1250) — compile-verified
//
#include <hip/hip_runtime.h>

// Problem dims (fixed by the reference)
#define Vn   50000
#define Bn   64
#define Sn   128
#define Hn   1024
#define Ln   3

typedef __attribute__((ext_vector_type(2))) float v2f;
typedef __attribute__((ext_vector_type(8))) float v8f;

// ---------------------------------------------------------------------------
// Multi-segment NT GEMM on the f32 WMMA pipe:
//   D[m,n] = sum_seg sum_k A_seg[m*lda+k] * W_seg[n*ldw_seg+k]  + bias1[n] + bias2[n]
// (segments with A==nullptr are skipped; all segments share lda and K)
//
// One wave computes a 64x16 strip (4 M-tiles of 16 share one B fragment);
// 4 waves/block -> 64x64 tile. M % 64 == 0, K % 4 == 0, N % 16 == 0
// (waves past N exit wave-uniformly so EXEC stays all-ones for WMMA).
//
// Addressing: uniform SGPR base + 32-bit *byte* offset (char* + zext) so the
// backend selects the saddr global-load form with immediate offsets instead
// of 64-bit per-lane address math. Inner loop is software-pipelined one
// k-step deep. W_NT=true marks single-use weight streams non-temporal
// (don't thrash the 192MB L2 with a 205MB out_W stream).
// ---------------------------------------------------------------------------
template <bool W_NT>
__global__ __launch_bounds__(128)
void gemm_nt_wmma_f32(const float* __restrict__ A0, const float* __restrict__ A1,
                      const float* __restrict__ A2, int lda,
                      const float* __restrict__ W0, const float* __restrict__ W1,
                      const float* __restrict__ W2,
                      int ldw0, int ldw1, int ldw2,
                      float* __restrict__ D, int ldd,
                      int M, int N, int K,
                      const float* __restrict__ bias1,
                      const float* __restrict__ bias2)
{
    const int lane = threadIdx.x & 31;
    const int wave = threadIdx.x >> 5;
    const int n0   = blockIdx.x * 64 + wave * 16;
    if (n0 >= N) return;                 // wave-uniform
    const int m_base = blockIdx.y * 64;
    const int nl = lane & 15;            // N (and M-row) index within tile
    const int kh = lane >> 4;            // selects K pair {0,1} vs {2,3}
    const unsigned kbh = 8u * (unsigned)kh;              // byte offset of K pair
    const unsigned ldaB = 4u * (unsigned)lda;            // row stride in bytes

    v8f acc0 = {}; v8f acc1 = {}; v8f acc2 = {}; v8f acc3 = {};

    const float* As[3]  = {A0, A1, A2};
    const float* Ws[3]  = {W0, W1, W2};
    const int    ldws[3] = {ldw0, ldw1, ldw2};

    const unsigned KB = 4u * (unsigned)K;                // bytes per row span

    for (int seg = 0; seg < 3; ++seg) {
        const float* A = As[seg];
        if (!A) break;
        const float* W = Ws[seg];
        const unsigned ldwB = 4u * (unsigned)ldws[seg];

        // 32-bit byte offsets from the uniform bases (max ~205MB -> fits u32/i32+)
        const unsigned wob  = (unsigned)(n0 + nl) * ldwB + kbh;
        const unsigned aob0 = (unsigned)(m_base + nl) * ldaB + kbh;
        const unsigned aob1 = aob0 + 16u * ldaB;
        const unsigned aob2 = aob0 + 32u * ldaB;
        const unsigned aob3 = aob0 + 48u * ldaB;

        auto ldW2 = [&](unsigned boff) -> v2f {
            const v2f* p = (const v2f*)((const char*)W + boff);
            if (W_NT) return __builtin_nontemporal_load(p);
            return *p;
        };
        auto ldA2 = [&](unsigned boff) -> v2f {
            return *(const v2f*)((const char*)A + boff);
        };

        // Prologue: fragments for k = 0
        v2f bf = ldW2(wob);
        v2f a0 = ldA2(aob0), a1 = ldA2(aob1), a2 = ldA2(aob2), a3 = ldA2(aob3);

        #pragma unroll 4
        for (unsigned kb = 16; kb < KB; kb += 16) {      // 16B = one k-step of 4 floats
            // issue next-iteration loads first (overlap with WMMAs below)
            v2f bfn = ldW2(wob  + kb);
            v2f a0n = ldA2(aob0 + kb);
            v2f a1n = ldA2(aob1 + kb);
            v2f a2n = ldA2(aob2 + kb);
            v2f a3n = ldA2(aob3 + kb);
            acc0 = __builtin_amdgcn_wmma_f32_16x16x4_f32(false, a0, false, bf, (short)0, acc0, false, false);
            acc1 = __builtin_amdgcn_wmma_f32_16x16x4_f32(false, a1, false, bf, (short)0, acc1, false, false);
            acc2 = __builtin_amdgcn_wmma_f32_16x16x4_f32(false, a2, false, bf, (short)0, acc2, false, false);
            acc3 = __builtin_amdgcn_wmma_f32_16x16x4_f32(false, a3, false, bf, (short)0, acc3, false, false);
            bf = bfn; a0 = a0n; a1 = a1n; a2 = a2n; a3 = a3n;
        }
        // Epilogue: last k-step of this segment
        acc0 = __builtin_amdgcn_wmma_f32_16x16x4_f32(false, a0, false, bf, (short)0, acc0, false, false);
        acc1 = __builtin_amdgcn_wmma_f32_16x16x4_f32(false, a1, false, bf, (short)0, acc1, false, false);
        acc2 = __builtin_amdgcn_wmma_f32_16x16x4_f32(false, a2, false, bf, (short)0, acc2, false, false);
        acc3 = __builtin_amdgcn_wmma_f32_16x16x4_f32(false, a3, false, bf, (short)0, acc3, false, false);
    }

    const int n = n0 + nl;
    float bias = 0.f;
    if (bias1) bias += bias1[n];
    if (bias2) bias += bias2[n];

    v8f accs[4] = {acc0, acc1, acc2, acc3};
    // C/D layout: VGPR r -> lanes0-15 (M=r, N=lane), lanes16-31 (M=r+8, N=lane-16)
    #pragma unroll
    for (int t = 0; t < 4; ++t) {
        #pragma unroll
        for (int r = 0; r < 8; ++r) {
            const int m = m_base + t * 16 + r + 8 * kh;
            const unsigned off = ((unsigned)m * (unsigned)ldd + (unsigned)n) * 4u;
            *(float*)((char*)D + off) = accs[t][r] + bias;
        }
    }
}

// ---------------------------------------------------------------------------
// x = emb[src]  (gather, [B,H])
// ---------------------------------------------------------------------------
__global__ void embed_kernel(const int* __restrict__ src,
                             const float* __restrict__ emb,
                             float* __restrict__ x)
{
    const int idx = blockIdx.x * blockDim.x + threadIdx.x;   // B*H total
    const int b = idx / Hn;
    const int j = idx - b * Hn;
    x[idx] = emb[(size_t)src[b] * Hn + j];
}

// ---------------------------------------------------------------------------
// Layer-0 attention tail: scores -> softmax(S) -> weighted sum over enc.
// One block per batch row b. e_a: [S,B,H] = enc@Wa^T ; e_hc: [B,H]
// ---------------------------------------------------------------------------
__global__ __launch_bounds__(256)
void attn_finalize_kernel(const float* __restrict__ e_a,
                          const float* __restrict__ e_hc,
                          const float* __restrict__ v0,
                          const float* __restrict__ enc,
                          float* __restrict__ weighted)
{
    __shared__ float s_ehc[Hn];
    __shared__ float s_v[Hn];
    __shared__ float s_sc[Sn];

    const int b    = blockIdx.x;
    const int tid  = threadIdx.x;
    const int lane = tid & 31;
    const int wv   = tid >> 5;

    for (int j = tid; j < Hn; j += 256) {
        s_ehc[j] = e_hc[b * Hn + j];
        s_v[j]   = v0[j];
    }
    __syncthreads();

    // scores[s] = sum_j tanh(e_a[s,b,j] + e_hc[b,j]) * v[j]; 8 waves, 16 s each
    for (int s = wv; s < Sn; s += 8) {
        const float* ea = e_a + ((size_t)s * Bn + b) * Hn;
        float acc = 0.f;
        for (int j = lane; j < Hn; j += 32)
            acc += tanhf(ea[j] + s_ehc[j]) * s_v[j];
        #pragma unroll
        for (int off = 16; off; off >>= 1) acc += __shfl_xor(acc, off, 32);
        if (lane == 0) s_sc[s] = acc;
    }
    __syncthreads();

    // softmax over S=128 done by wave 0 (4 values per lane)
    if (wv == 0) {
        float vals[4];
        float mx = -3.4e38f;
        #pragma unroll
        for (int i = 0; i < 4; ++i) { vals[i] = s_sc[lane + 32 * i]; mx = fmaxf(mx, vals[i]); }
        #pragma unroll
        for (int off = 16; off; off >>= 1) mx = fmaxf(mx, __shfl_xor(mx, off, 32));
        float sum = 0.f;
        #pragma unroll
        for (int i = 0; i < 4; ++i) { vals[i] = __expf(vals[i] - mx); sum += vals[i]; }
        #pragma unroll
        for (int off = 16; off; off >>= 1) sum += __shfl_xor(sum, off, 32);
        const float inv = 1.f / sum;
        #pragma unroll
        for (int i = 0; i < 4; ++i) s_sc[lane + 32 * i] = vals[i] * inv;
    }
    __syncthreads();

    // weighted[b,h] = sum_s a[s] * enc[s,b,h]
    for (int h = tid; h < Hn; h += 256) {
        float acc = 0.f;
        for (int s = 0; s < Sn; ++s)
            acc += s_sc[s] * enc[((size_t)s * Bn + b) * Hn + h];
        weighted[b * Hn + h] = acc;
    }
}

// ---------------------------------------------------------------------------
// LSTM pointwise: gates [B,4H] (i,f,g,o), c_old [B,H] -> h_new, c_new [B,H]
// ---------------------------------------------------------------------------
__device__ __forceinline__ float sigmoidf_(float x) { return 1.f / (1.f + __expf(-x)); }

__global__ void lstm_pointwise_kernel(const float* __restrict__ gates,
                                      const float* __restrict__ c_old,
                                      float* __restrict__ h_new,
                                      float* __restrict__ c_new)
{
    const int idx = blockIdx.x * blockDim.x + threadIdx.x;   // B*H total
    const int b = idx / Hn;
    const int j = idx - b * Hn;
    const float* g = gates + (size_t)b * 4 * Hn;
    const float i_ = g[j];
    const float f_ = g[Hn + j];
    const float g_ = g[2 * Hn + j];
    const float o_ = g[3 * Hn + j];
    const float c2 = sigmoidf_(f_) * c_old[idx] + sigmoidf_(i_) * tanhf(g_);
    c_new[idx] = c2;
    h_new[idx] = sigmoidf_(o_) * tanhf(c2);
}

// ---------------------------------------------------------------------------
// In-place row log-softmax over N=V per block row.
// ---------------------------------------------------------------------------
__global__ __launch_bounds__(256)
void log_softmax_kernel(float* __restrict__ logits, int N)
{
    __shared__ float red[8];
    const int b = blockIdx.x;
    float* row = logits + (size_t)b * N;
    const int tid = threadIdx.x, lane = tid & 31, wv = tid >> 5;

    float m = -3.4e38f;
    for (int v = tid; v < N; v += 256) m = fmaxf(m, row[v]);
    #pragma unroll
    for (int off = 16; off; off >>= 1) m = fmaxf(m, __shfl_xor(m, off, 32));
    if (lane == 0) red[wv] = m;
    __syncthreads();
    m = red[0];
    #pragma unroll
    for (int i = 1; i < 8; ++i) m = fmaxf(m, red[i]);
    __syncthreads();

    float s = 0.f;
    for (int v = tid; v < N; v += 256) s += __expf(row[v] - m);
    #pragma unroll
    for (int off = 16; off; off >>= 1) s += __shfl_xor(s, off, 32);
    if (lane == 0) red[wv] = s;
    __syncthreads();
    s = 0.f;
    #pragma unroll
    for (int i = 0; i < 8; ++i) s += red[i];

    const float lse = m + __logf(s);
    for (int v = tid; v < N; v += 256) row[v] = row[v] - lse;
}

// ---------------------------------------------------------------------------
extern "C" void kernel_launch(void* const* d_in, const int* in_sizes, int n_in,
                              void* d_out, int out_size, void* d_ws, size_t ws_size,
                              hipStream_t stream)
{
    const int*   src    = (const int*)  d_in[0];
    const float* hidden = (const float*)d_in[1];   // [L,B,H]
    const float* cell   = (const float*)d_in[2];   // [L,B,H]
    const float* enc    = (const float*)d_in[3];   // [S,B,H]
    const float* emb    = (const float*)d_in[4];   // [V,H]
    const float* attn_W = (const float*)d_in[5];   // [H,3H]
    const float* attn_b = (const float*)d_in[6];   // [H]
    const float* attn_v = (const float*)d_in[7];   // [L,H] (only row 0 matters)
    const float* W_ih[Ln] = {(const float*)d_in[8],  (const float*)d_in[12], (const float*)d_in[16]};
    const float* W_hh[Ln] = {(const float*)d_in[9],  (const float*)d_in[13], (const float*)d_in[17]};
    const float* b_ih[Ln] = {(const float*)d_in[10], (const float*)d_in[14], (const float*)d_in[18]};
    const float* b_hh[Ln] = {(const float*)d_in[11], (const float*)d_in[15], (const float*)d_in[19]};
    const float* out_W  = (const float*)d_in[20];  // [V,H]
    const float* out_b  = (const float*)d_in[21];  // [V]

    float* out    = (float*)d_out;
    float* logits = out;                                    // [B,V]
    float* hs     = out + (size_t)Bn * Vn;                  // [L,B,H]
    float* cs     = hs  + (size_t)Ln * Bn * Hn;             // [L,B,H]

    float* ws       = (float*)d_ws;
    float* x        = ws;                  // [B,H]
    float* ehc      = ws + 1 * Bn * Hn;    // [B,H]
    float* weighted = ws + 2 * Bn * Hn;    // [B,H]
    float* gates    = ws + 3 * Bn * Hn;    // [B,4H]
    float* ea       = ws + 7 * Bn * Hn;    // [S,B,H]

    // Up-to-3-segment fused GEMM; nt=true -> weight streams non-temporal.
    auto gemm3 = [&](const float* A0, const float* A1, const float* A2, int lda,
                     const float* W0, const float* W1, const float* W2,
                     int ldw0, int ldw1, int ldw2,
                     float* Dm, int ldd, int M, int N, int K,
                     const float* bb1, const float* bb2, bool nt) {
        dim3 g((N + 63) / 64, M / 64);
        if (nt)
            gemm_nt_wmma_f32<true><<<g, 128, 0, stream>>>(A0, A1, A2, lda, W0, W1, W2,
                                                          ldw0, ldw1, ldw2,
                                                          Dm, ldd, M, N, K, bb1, bb2);
        else
            gemm_nt_wmma_f32<false><<<g, 128, 0, stream>>>(A0, A1, A2, lda, W0, W1, W2,
                                                           ldw0, ldw1, ldw2,
                                                           Dm, ldd, M, N, K, bb1, bb2);
    };

    // x = emb[src]
    embed_kernel<<<(Bn * Hn) / 256, 256, 0, stream>>>(src, emb, x);

    // --- layer-0 attention (layers 1/2: softmax over S=1 == 1 -> weighted = h_prev) ---
    // e_hc = h0 @ Wh^T + c0 @ Wc^T + attn_b     (one fused launch, 2 segments)
    gemm3(hidden, cell, nullptr, Hn,
          attn_W, attn_W + Hn, nullptr, 3 * Hn, 3 * Hn, 0,
          ehc, Hn, Bn, Hn, Hn, attn_b, nullptr, false);
    // e_a = enc(flattened [S*B,H]) @ Wa^T       (W slab re-read by 128 M-blocks -> RT)
    gemm3(enc, nullptr, nullptr, Hn,
          attn_W + 2 * Hn, nullptr, nullptr, 3 * Hn, 0, 0,
          ea, Hn, Sn * Bn, Hn, Hn, nullptr, nullptr, false);
    attn_finalize_kernel<<<Bn, 256, 0, stream>>>(ea, ehc, attn_v, enc, weighted);

    // --- 3 stacked LSTM layers: one fused 3-segment GEMM per layer (weights NT) ---
    const float* xin = x;
    const float* win = weighted;
    for (int i = 0; i < Ln; ++i) {
        const float* h_i = hidden + (size_t)i * Bn * Hn;
        const float* c_i = cell   + (size_t)i * Bn * Hn;
        // gates = x @ Wih[:, :H]^T + weighted @ Wih[:, H:]^T + h_i @ Whh^T + b_ih + b_hh
        gemm3(xin, win, h_i, Hn,
              W_ih[i], W_ih[i] + Hn, W_hh[i], 2 * Hn, 2 * Hn, Hn,
              gates, 4 * Hn, Bn, 4 * Hn, Hn, b_ih[i], b_hh[i], true);
        float* h_o = hs + (size_t)i * Bn * Hn;
        float* c_o = cs + (size_t)i * Bn * Hn;
        lstm_pointwise_kernel<<<(Bn * Hn) / 256, 256, 0, stream>>>(gates, c_i, h_o, c_o);
        xin = h_o;   // x = h_prev
        win = h_o;   // weighted = h_prev (singleton softmax == 1)
    }

    // logits = h2 @ out_W^T + out_b (205MB single-use stream -> NT); then log-softmax
    gemm3(xin, nullptr, nullptr, Hn,
          out_W, nullptr, nullptr, Hn, 0, 0,
          logits, Vn, Bn, Vn, Hn, out_b, nullptr, true);
    log_softmax_kernel<<<Bn, 256, 0, stream>>>(logits, Vn);
}